// InteractionModule_14491219657363
// MI455X (gfx1250) — compile-verified
//
#include <hip/hip_runtime.h>
#include <hip/hip_bf16.h>
#include <stdint.h>

typedef __attribute__((ext_vector_type(2))) float v2f;
typedef __attribute__((ext_vector_type(4))) float v4f;
typedef __attribute__((ext_vector_type(8))) float v8f;

#define BATCH 2
#define NP 512
#define NC 512
#define DD 128
#define PP 128
#define LDA 132      // padded LDS row stride (floats): banks spaced by 4, rows 16B aligned
#define SMEM_BYTES (2 * 128 * LDA * 4)

#if defined(__HIP_DEVICE_COMPILE__)
#if !__has_builtin(__builtin_amdgcn_wmma_f32_16x16x4_f32)
#error "missing __builtin_amdgcn_wmma_f32_16x16x4_f32 on device toolchain"
#endif
#endif

// ---------------- LayerNorm pre-pass: one wave (32 lanes) per row of D=128 ----------------
__global__ __launch_bounds__(256) void ln_rows_kernel(const float* __restrict__ x,
                                                      const float* __restrict__ w,
                                                      const float* __restrict__ bvec,
                                                      float* __restrict__ y) {
    int wave = threadIdx.x >> 5;
    int lane = threadIdx.x & 31;
    int row  = blockIdx.x * 8 + wave;
    const float* xr = x + (size_t)row * DD;
    v4f v = *(const v4f*)(xr + lane * 4);

    float s = v.x + v.y + v.z + v.w;
    #pragma unroll
    for (int off = 16; off > 0; off >>= 1) s += __shfl_xor(s, off, 32);
    float mu = s * (1.0f / DD);

    v4f d = v - mu;
    float q = d.x * d.x + d.y * d.y + d.z * d.z + d.w * d.w;
    #pragma unroll
    for (int off = 16; off > 0; off >>= 1) q += __shfl_xor(q, off, 32);
    float rstd = __frsqrt_rn(q * (1.0f / DD) + 1e-5f);

    v4f wv = *(const v4f*)(w + lane * 4);
    v4f bb = *(const v4f*)(bvec + lane * 4);
    v4f o  = d * rstd * wv + bb;
    *(v4f*)(y + (size_t)row * DD + lane * 4) = o;
}

// ---------------- Main fused interaction GEMM ----------------
// Block = 256 threads (8 waves, 4(j) x 2(h) wave grid). One block computes a
// 128(j) x 128(h) tile of out[b,i,:,:] for one (b, i, j-tile).
// A[j,k] = cn[b,j,k] * pn[b,i,k]; B = W_out (staged via async global->LDS DMA).
__global__ __launch_bounds__(256) void interact_wmma_kernel(
    const float* __restrict__ pn,        // [B,NP,D] normalized p
    const float* __restrict__ cn,        // [B,NC,D] normalized c
    const unsigned char* __restrict__ p_mask,  // [B,NP]
    const unsigned char* __restrict__ c_mask,  // [B,NC]
    const float* __restrict__ Wout,      // [P,D] row-major
    const float* __restrict__ bout,      // [P]
    float* __restrict__ out,             // [B,NP,NC,P]
    float* __restrict__ mask_out)        // [B,NP,NC] as float 0/1
{
    extern __shared__ float smem[];
    float* As = smem;             // 128 x LDA  (A tile, row-major j x k)
    float* Bs = smem + 128 * LDA; // 128 x LDA  (W rows, h x k)

    int blk = blockIdx.x;
    int jt  = blk & 3;
    int i   = (blk >> 2) & (NP - 1);
    int b   = blk >> 11;

    int t    = threadIdx.x;
    int kcol = (t & 31) * 4;   // k column group
    int rsub = t >> 5;         // 0..7

    // p-row fragment (same k-columns for all staged rows)
    v4f pv = *(const v4f*)(pn + (((size_t)b * NP + i) * DD) + kcol);
    const float* cbase = cn + (((size_t)b * NC + (size_t)jt * 128) * DD);

#if defined(__HIP_DEVICE_COMPILE__)
    // Async DMA the W tile straight into LDS (ASYNCcnt path), overlap with
    // VALU staging of the scaled A tile.
    uint32_t bs_base = (uint32_t)(uintptr_t)Bs;  // LDS byte offset (flat-aperture low bits)
    #pragma unroll
    for (int pass = 0; pass < 16; ++pass) {
        int r = pass * 8 + rsub;
        uint32_t goff = (uint32_t)((r * DD + kcol) * 4);
        uint32_t loff = bs_base + (uint32_t)((r * LDA + kcol) * 4);
        asm volatile("global_load_async_to_lds_b128 %0, %1, %2"
                     :: "v"(loff), "v"(goff), "s"(Wout) : "memory");
        v4f cv = *(const v4f*)(cbase + (size_t)r * DD + kcol);
        *(v4f*)(&As[r * LDA + kcol]) = cv * pv;
    }
    asm volatile("s_wait_asynccnt 0x0" ::: "memory");
#else
    for (int pass = 0; pass < 16; ++pass) {
        int r = pass * 8 + rsub;
        v4f cv = *(const v4f*)(cbase + (size_t)r * DD + kcol);
        *(v4f*)(&As[r * LDA + kcol]) = cv * pv;
        v4f wv = *(const v4f*)(Wout + (size_t)r * DD + kcol);
        *(v4f*)(&Bs[r * LDA + kcol]) = wv;
    }
#endif
    __syncthreads();

    int wave = t >> 5;
    int lane = t & 31;
    int jw   = wave & 3;    // j wave group: rows jw*32 .. jw*32+31
    int hw   = wave >> 2;   // h wave group: cols hw*64 .. hw*64+63
    int m    = lane & 15;   // A: row M / B,C: col N
    int hi   = lane >> 4;   // half-wave selector (K offset +2, C row +8)

    // Accumulators seeded with bias: acc[mi*4+n] is 16x16 tile at
    // (j = jw*32 + mi*16, h = hw*64 + n*16).
    v8f acc[8];
    #pragma unroll
    for (int n = 0; n < 4; ++n) {
        float bb = bout[hw * 64 + n * 16 + m];
        #pragma unroll
        for (int v = 0; v < 8; ++v) { acc[n][v] = bb; acc[4 + n][v] = bb; }
    }

#if defined(__HIP_DEVICE_COMPILE__)
    const float* arow0 = &As[(jw * 32 + m) * LDA];
    const float* arow1 = &As[(jw * 32 + 16 + m) * LDA];
    const float* brow  = &Bs[(hw * 64 + m) * LDA];
    #pragma unroll 2
    for (int k = 0; k < DD; k += 4) {
        int kk = k + 2 * hi;
        v2f af0 = *(const v2f*)(arow0 + kk);   // ds_load_b64: {K,K+1} per half-wave
        v2f af1 = *(const v2f*)(arow1 + kk);
        #pragma unroll
        for (int n = 0; n < 4; ++n) {
            v2f bf = *(const v2f*)(brow + n * 16 * LDA + kk);
            acc[n] = __builtin_amdgcn_wmma_f32_16x16x4_f32(
                false, af0, false, bf, (short)0, acc[n], false, false);
            acc[4 + n] = __builtin_amdgcn_wmma_f32_16x16x4_f32(
                false, af1, false, bf, (short)0, acc[4 + n], false, false);
        }
    }
#endif

    float pm = p_mask[(size_t)b * NP + i] ? 1.0f : 0.0f;
    size_t outBase = (((size_t)b * NP + i) * NC + (size_t)jt * 128) * PP;

    #pragma unroll
    for (int mi = 0; mi < 2; ++mi) {
        #pragma unroll
        for (int v = 0; v < 8; ++v) {
            int j = jw * 32 + mi * 16 + hi * 8 + v;   // C layout: VGPR v -> M = v + 8*hi
            float cm = c_mask[(size_t)b * NC + (size_t)jt * 128 + j] ? 1.0f : 0.0f;
            float mk = pm * cm;
            #pragma unroll
            for (int n = 0; n < 4; ++n) {
                __builtin_nontemporal_store(
                    acc[mi * 4 + n][v] * mk,
                    out + outBase + (size_t)j * PP + hw * 64 + n * 16 + m);
            }
        }
    }

    // mask output (one unique (b,i,j-tile) per block)
    if (t < 128) {
        int j = jt * 128 + t;
        float cm = c_mask[(size_t)b * NC + j] ? 1.0f : 0.0f;
        __builtin_nontemporal_store(pm * cm,
            mask_out + (((size_t)b * NP + i) * NC + j));
    }
}

extern "C" void kernel_launch(void* const* d_in, const int* in_sizes, int n_in,
                              void* d_out, int out_size, void* d_ws, size_t ws_size,
                              hipStream_t stream) {
    const float* p_embed = (const float*)d_in[0];
    const float* c_embed = (const float*)d_in[1];
    const unsigned char* p_mask = (const unsigned char*)d_in[2];
    const unsigned char* c_mask = (const unsigned char*)d_in[3];
    const float* ln_p_w = (const float*)d_in[4];
    const float* ln_p_b = (const float*)d_in[5];
    const float* ln_c_w = (const float*)d_in[6];
    const float* ln_c_b = (const float*)d_in[7];
    const float* W_out  = (const float*)d_in[8];
    const float* b_out  = (const float*)d_in[9];

    float* pn = (float*)d_ws;                       // [B,NP,D]
    float* cn = pn + (size_t)BATCH * NP * DD;       // [B,NC,D]

    ln_rows_kernel<<<BATCH * NP / 8, 256, 0, stream>>>(p_embed, ln_p_w, ln_p_b, pn);
    ln_rows_kernel<<<BATCH * NC / 8, 256, 0, stream>>>(c_embed, ln_c_w, ln_c_b, cn);

    float* out      = (float*)d_out;
    float* mask_out = out + (size_t)BATCH * NP * NC * PP;

    interact_wmma_kernel<<<BATCH * NP * (NC / 128), 256, SMEM_BYTES, stream>>>(
        pn, cn, p_mask, c_mask, W_out, b_out, out, mask_out);
}